// SSIMLoss_36060545417278
// MI455X (gfx1250) — compile-verified
//
#include <hip/hip_runtime.h>
#include <math.h>
#include <stdint.h>

// ---------------- problem constants (shape fixed by setup_inputs) ----------
#define IMG        512
#define CHN        3
#define BATCH      32
#define KWIN       11
#define TILE       16
#define IN_ROWS    26              // TILE + KWIN - 1
#define LDS_COLS   32              // aligned staging width (cols tx*16-8 .. +23)
#define COL_SHIFT  3               // logical conv col 0 lives at LDS col 3
#define H_ROWS     28              // 26 + 2 zero rows (K padding)
#define TILES_1D   (IMG / TILE)                         // 32
#define TILES_TOT  (BATCH * CHN * TILES_1D * TILES_1D)  // 98304
#define WPB        6               // waves per block (192 threads)
#define NBLOCKS    (TILES_TOT / WPB)                    // 16384
#define SMEM_PER_WAVE (IN_ROWS*LDS_COLS*2 + H_ROWS*TILE) // 2112 floats
#define C1V 0.0001f   // 0.01^2
#define C2V 0.0009f   // 0.03^2

typedef float v2f __attribute__((ext_vector_type(2)));
typedef float v8f __attribute__((ext_vector_type(8)));

__device__ __forceinline__ v8f wmma4_f32(v2f a, v2f b, v8f c) {
  // D(16x16,f32) = A(16x4,f32) x B(4x16,f32) + C   -> v_wmma_f32_16x16x4_f32
  return __builtin_amdgcn_wmma_f32_16x16x4_f32(false, a, false, b,
                                               (short)0, c, false, false);
}

__global__ __launch_bounds__(192)
void ssim_tile_kernel(const float* __restrict__ img1,
                      const float* __restrict__ img2,
                      float* __restrict__ partials) {
  __shared__ __align__(16) float smem[WPB * SMEM_PER_WAVE];
  __shared__ float gw[KWIN];
  __shared__ float red[WPB];

  const int tid  = threadIdx.x;
  const int lane = tid & 31;
  const int wv   = tid >> 5;

  // Gaussian taps (sigma = 1.5), normalized
  if (tid == 0) {
    float tmp[KWIN];
    float s = 0.f;
#pragma unroll
    for (int i = 0; i < KWIN; ++i) {
      float x = (float)(i - KWIN / 2);
      tmp[i] = expf(-(x * x) / (2.f * 1.5f * 1.5f));
      s += tmp[i];
    }
#pragma unroll
    for (int i = 0; i < KWIN; ++i) gw[i] = tmp[i] / s;
  }
  __syncthreads();

  // ---- per-wave tile assignment --------------------------------------
  const int t     = blockIdx.x * WPB + wv;       // 0 .. TILES_TOT-1
  const int tx    = t & (TILES_1D - 1);
  const int ty    = (t >> 5) & (TILES_1D - 1);
  const int plane = t >> 10;                     // 0 .. BATCH*CHN-1
  const float* p1 = img1 + (size_t)plane * (IMG * IMG);
  const float* p2 = img2 + (size_t)plane * (IMG * IMG);
  const int oy    = ty * TILE - (KWIN / 2);      // first halo row
  const int ox_al = tx * TILE - 8;               // aligned first staged col

  float* sI1 = smem + wv * SMEM_PER_WAVE;
  float* sI2 = sI1 + IN_ROWS * LDS_COLS;
  float* sH  = sI2 + IN_ROWS * LDS_COLS;

  // ---- stage halo tiles into LDS -------------------------------------
  const bool interior = (tx >= 1) && (tx <= 30) && (ty >= 1) && (ty <= 30);
  if (interior) {
    // fully in-bounds: async 128-bit global->LDS DMA (ASYNCcnt path)
    const uint64_t b1 = (uint64_t)(uintptr_t)p1;
    const uint64_t b2 = (uint64_t)(uintptr_t)p2;
    for (int e = lane; e < IN_ROWS * (LDS_COLS / 4); e += 32) {  // 208 xfers
      int rr = e >> 3, c4 = (e & 7) << 2;
      unsigned goff = (unsigned)((((oy + rr) * IMG) + ox_al + c4) * 4);
      unsigned l1 = (unsigned)(uintptr_t)(void*)&sI1[rr * LDS_COLS + c4];
      unsigned l2 = (unsigned)(uintptr_t)(void*)&sI2[rr * LDS_COLS + c4];
      asm volatile("global_load_async_to_lds_b128 %0, %1, %2 offset:0"
                   :: "v"(l1), "v"(goff), "s"(b1) : "memory");
      asm volatile("global_load_async_to_lds_b128 %0, %1, %2 offset:0"
                   :: "v"(l2), "v"(goff), "s"(b2) : "memory");
    }
    asm volatile("s_wait_asynccnt 0x0" ::: "memory");
  } else {
    // border: scalar loads with zero padding (matches reference conv pad)
    for (int e = lane; e < IN_ROWS * LDS_COLS; e += 32) {
      int rr = e >> 5, cc = e & 31;
      int gy = oy + rr, gx = ox_al + cc;
      bool ok = ((unsigned)gy < (unsigned)IMG) && ((unsigned)gx < (unsigned)IMG);
      int gi = gy * IMG + gx;
      sI1[e] = ok ? p1[gi] : 0.f;
      sI2[e] = ok ? p2[gi] : 0.f;
    }
  }
  // K-padding rows 26,27 of H are never written by the passes -> zero once
  sH[26 * TILE + lane] = 0.f;   // 32 entries = rows 26..27 x 16 cols

  const int hs   = (lane < 16) ? 0 : 2;   // K half-select for A/B operands
  const int nsub = lane & 15;             // N / M sub index

  // Band weights are identical for horizontal-B and vertical-A operands:
  //   w(s)[d] = g[(4s+hs+d) - nsub] on the band, else 0
  v2f WG[7];
#pragma unroll
  for (int s = 0; s < 7; ++s) {
    int j = 4 * s + hs;
    unsigned d0 = (unsigned)(j - nsub);
    unsigned d1 = (unsigned)(j + 1 - nsub);
    float w0 = gw[d0 < 11u ? d0 : 0u];  if (d0 >= 11u) w0 = 0.f;
    float w1 = gw[d1 < 11u ? d1 : 0u];  if (d1 >= 11u) w1 = 0.f;
    WG[s].x = w0; WG[s].y = w1;
  }

  v8f vres[5];
#pragma unroll
  for (int q = 0; q < 5; ++q) {
    // ---------- horizontal pass: H(26x16) = Q(26x28) x Wh(28x16) ------
    v8f hacc[2];
#pragma unroll
    for (int rb = 0; rb < 2; ++rb) {
      v8f acc = {0.f,0.f,0.f,0.f,0.f,0.f,0.f,0.f};
      int m = nsub + rb * 16;
      if (m > 25) m = 25;                       // clamp garbage rows (w=0 later)
      const float* r1 = sI1 + m * LDS_COLS + COL_SHIFT;
      const float* r2 = sI2 + m * LDS_COLS + COL_SHIFT;
#pragma unroll
      for (int s = 0; s < 7; ++s) {
        int k = 4 * s + hs;
        float a10 = r1[k], a11 = r1[k + 1];
        float a20 = r2[k], a21 = r2[k + 1];
        float q0, q1;
        if      (q == 0) { q0 = a10;       q1 = a11;       }  // img1
        else if (q == 1) { q0 = a20;       q1 = a21;       }  // img2
        else if (q == 2) { q0 = a10 * a10; q1 = a11 * a11; }  // img1^2
        else if (q == 3) { q0 = a20 * a20; q1 = a21 * a21; }  // img2^2
        else             { q0 = a10 * a20; q1 = a11 * a21; }  // img1*img2
        v2f A = {q0, q1};
        acc = wmma4_f32(A, WG[s], acc);
      }
      hacc[rb] = acc;
    }
    // C layout (VGPR i -> rows i / i+8) -> LDS rows of H (B-operand order)
#pragma unroll
    for (int i = 0; i < 8; ++i) {
      int r0 = i + ((lane < 16) ? 0 : 8);
      sH[r0 * TILE + nsub] = hacc[0][i];
      int r1r = 16 + r0;
      if (r1r < 26) sH[r1r * TILE + nsub] = hacc[1][i];
    }
    // ---------- vertical pass: O(16x16) = Wv(16x28) x H(28x16) --------
    v8f vac = {0.f,0.f,0.f,0.f,0.f,0.f,0.f,0.f};
#pragma unroll
    for (int s = 0; s < 7; ++s) {
      int j = 4 * s + hs;
      v2f B = { sH[j * TILE + nsub], sH[(j + 1) * TILE + nsub] };
      vac = wmma4_f32(WG[s], B, vac);
    }
    vres[q] = vac;
  }

  // ---------------- SSIM map + deterministic reduction -----------------
  float lsum = 0.f;
#pragma unroll
  for (int i = 0; i < 8; ++i) {
    float mu1 = vres[0][i], mu2 = vres[1][i];
    float mu1s = mu1 * mu1, mu2s = mu2 * mu2, m12 = mu1 * mu2;
    float s1  = vres[2][i] - mu1s;
    float s2  = vres[3][i] - mu2s;
    float s12 = vres[4][i] - m12;
    float num = (2.f * m12 + C1V) * (2.f * s12 + C2V);
    float den = (mu1s + mu2s + C1V) * (s1 + s2 + C2V);
    lsum += num / den;
  }
  for (int off = 16; off > 0; off >>= 1)
    lsum += __shfl_down(lsum, off, 32);
  if (lane == 0) red[wv] = lsum;
  __syncthreads();
  if (tid == 0) {
    float tsum = 0.f;
#pragma unroll
    for (int w = 0; w < WPB; ++w) tsum += red[w];
    partials[blockIdx.x] = tsum;
  }
}

__global__ __launch_bounds__(256)
void ssim_reduce_kernel(const float* __restrict__ partials,
                        float* __restrict__ out) {
  __shared__ float r[256];
  int tid = threadIdx.x;
  float s = 0.f;
  for (int i = tid; i < NBLOCKS; i += 256) s += partials[i];
  r[tid] = s;
  __syncthreads();
  for (int st = 128; st > 0; st >>= 1) {
    if (tid < st) r[tid] += r[tid + st];
    __syncthreads();
  }
  if (tid == 0)
    out[0] = 1.0f - r[0] / (float)((size_t)BATCH * CHN * IMG * IMG);
}

extern "C" void kernel_launch(void* const* d_in, const int* in_sizes, int n_in,
                              void* d_out, int out_size, void* d_ws, size_t ws_size,
                              hipStream_t stream) {
  const float* img1 = (const float*)d_in[0];
  const float* img2 = (const float*)d_in[1];
  float* partials   = (float*)d_ws;      // NBLOCKS floats (64 KB) of scratch
  float* out        = (float*)d_out;     // single fp32 scalar

  ssim_tile_kernel<<<NBLOCKS, 192, 0, stream>>>(img1, img2, partials);
  ssim_reduce_kernel<<<1, 256, 0, stream>>>(partials, out);
}